// Model_49194555408902
// MI455X (gfx1250) — compile-verified
//
#include <hip/hip_runtime.h>
#include <math.h>

typedef __attribute__((ext_vector_type(2))) float v2f;
typedef __attribute__((ext_vector_type(8))) float v8f;

#define FLAG_TRANSB  1
#define FLAG_BIAS    2
#define FLAG_LEAKY   4
#define FLAG_CLUSTER 8

__device__ __forceinline__ float sigf(float x) { return 1.f / (1.f + expf(-x)); }

// ---------------------------------------------------------------------------
// Generic batched WMMA GEMM (fp32, V_WMMA_F32_16X16X4_F32).
// C[b,o][m,n] = sum_k A[b,o][m,k] * B'[k,n] (+bias[n]) (leaky / +marg@Wc epilogues)
// B' = B or B^T depending on FLAG_TRANSB. One 16x16 tile per wave, 8 waves/block.
// Out-of-range M/N lanes are CLAMPED (not predicated): the garbage they load
// only reaches accumulator elements that are never stored, so the K main loop
// is branch-free: pointer-increment loads feeding v_wmma.
// ---------------------------------------------------------------------------
__global__ void gemm_wmma(const float* __restrict__ A, const float* __restrict__ B,
                          const float* __restrict__ bias, float* __restrict__ C,
                          const float* __restrict__ Wc,
                          int M, int N, int K,
                          int lda, int ldb, int ldc, int stc,
                          long aSb, long aSo, long bSb, long bSo,
                          long cSb, long cSo, int biasSo,
                          int flags, int oBase)
{
    const int lane = threadIdx.x & 31;
    const int wave = threadIdx.x >> 5;
    const int l16  = lane & 15;
    const int hl   = lane >> 4;           // 0: lanes 0-15, 1: lanes 16-31
    const int tilesN = (N + 15) >> 4;
    const int tilesM = (M + 15) >> 4;
    int tile = blockIdx.x * (blockDim.x >> 5) + wave;
    if (tile >= tilesM * tilesN) return;  // wave-uniform exit (EXEC stays all-1)
    const int tm = tile / tilesN, tn = tile % tilesN;
    const int bb = blockIdx.y, oo = blockIdx.z;

    const float* Ap = A + bb * aSb + oo * aSo;
    const float* Bp = B + bb * bSb + oo * bSo;
    const int rowA = tm * 16 + l16;
    const int colB = tn * 16 + l16;
    const int rA = rowA < M ? rowA : M - 1;   // clamp: always-valid addresses
    const int cB = colB < N ? colB : N - 1;
    const bool trb = (flags & FLAG_TRANSB) != 0;
    const int Kmain = K & ~3;

    const float* ap = Ap + (long)rA * lda + 2 * hl;
    const float* bp = trb ? (Bp + (long)cB * ldb + 2 * hl)
                          : (Bp + (long)(2 * hl) * ldb + cB);
    const long bstep = trb ? 4 : 4L * ldb;
    const long byoff = trb ? 1 : ldb;     // offset between the 2 B K-rows

    v8f acc = {0.f, 0.f, 0.f, 0.f, 0.f, 0.f, 0.f, 0.f};
    if (Kmain > 0) {
        v2f a, b;
        a.x = ap[0];     a.y = ap[1];
        b.x = bp[0];     b.y = bp[byoff];
        for (int k0 = 4; k0 < Kmain; k0 += 4) {
            ap += 4; bp += bstep;
            v2f an, bn;                        // prefetch next fragments
            an.x = ap[0];  an.y = ap[1];
            bn.x = bp[0];  bn.y = bp[byoff];
            acc = __builtin_amdgcn_wmma_f32_16x16x4_f32(false, a, false, b,
                                                        (short)0, acc, false, false);
            a = an; b = bn;
        }
        acc = __builtin_amdgcn_wmma_f32_16x16x4_f32(false, a, false, b,
                                                    (short)0, acc, false, false);
    }
    if (Kmain < K) {                      // K tail: branchless register selects
        const int ka = Kmain + 2 * hl, kb = ka + 1;
        const int kac = ka < K ? ka : K - 1;
        const int kbc = kb < K ? kb : K - 1;
        v2f a, b;
        a.x = Ap[(long)rA * lda + kac];
        a.y = Ap[(long)rA * lda + kbc];
        if (trb) {
            b.x = Bp[(long)cB * ldb + kac];
            b.y = Bp[(long)cB * ldb + kbc];
        } else {
            b.x = Bp[(long)kac * ldb + cB];
            b.y = Bp[(long)kbc * ldb + cB];
        }
        if (ka >= K) { a.x = 0.f; b.x = 0.f; }
        if (kb >= K) { a.y = 0.f; b.y = 0.f; }
        acc = __builtin_amdgcn_wmma_f32_16x16x4_f32(false, a, false, b,
                                                    (short)0, acc, false, false);
    }

    float* Cp = C + bb * cSb + oo * cSo;
    const bool full = (tm * 16 + 16 <= M) && (tn * 16 + 16 <= N); // wave-uniform
    const int nn = tn * 16 + l16;
    if (full) {
        for (int r = 0; r < 8; ++r) {
            const int m = tm * 16 + r + 8 * hl;
            float v = acc[r];
            if (flags & FLAG_BIAS)  v += bias[(long)oo * biasSo + nn];
            if (flags & FLAG_LEAKY) v = v > 0.f ? v : 0.1f * v;
            const long off = (long)m * ldc + (long)nn * stc;
            if (flags & FLAG_CLUSTER) {
                const int o = oBase + oo;
                const float* mp = Cp + off - (2 + o);   // marg[...,0:2] in d_out
                v += mp[0] * Wc[o * 2 + 0] + mp[1] * Wc[o * 2 + 1];
            }
            Cp[off] = v;
        }
    } else {
        for (int r = 0; r < 8; ++r) {
            const int m = tm * 16 + r + 8 * hl;
            if (m < M && nn < N) {
                float v = acc[r];
                if (flags & FLAG_BIAS)  v += bias[(long)oo * biasSo + nn];
                if (flags & FLAG_LEAKY) v = v > 0.f ? v : 0.1f * v;
                const long off = (long)m * ldc + (long)nn * stc;
                if (flags & FLAG_CLUSTER) {
                    const int o = oBase + oo;
                    const float* mp = Cp + off - (2 + o);
                    v += mp[0] * Wc[o * 2 + 0] + mp[1] * Wc[o * 2 + 1];
                }
                Cp[off] = v;
            }
        }
    }
}

// ---------------------------------------------------------------------------
// Embedding gather: emb[b*128+s, 0:300] = word_embed[words[b,s]]
// ---------------------------------------------------------------------------
__global__ void embed_kernel(const float* __restrict__ we, const int* __restrict__ words,
                             float* __restrict__ out)
{
    long i = (long)blockIdx.x * blockDim.x + threadIdx.x;
    if (i >= 2048L * 300) return;
    long r = i / 300, e = i % 300;
    out[i] = we[(long)words[r] * 300 + e];
}

// ---------------------------------------------------------------------------
// Persistent bidirectional LSTM recurrence. grid.x = direction (0 fwd, 1 bwd).
// 25 waves; wave w owns hidden units [16w,16w+16) across all 4 gates; the cell
// state lives in an accumulator fragment for the whole 128-step scan. h is
// shared through LDS with one barrier pair per step. All fragment loads are
// 8-byte aligned pairs -> v2f (b64) loads with per-gate pointer increments.
// ---------------------------------------------------------------------------
__global__ void lstm_scan(const float* __restrict__ xw, const float* __restrict__ whh,
                          float* __restrict__ out)
{
    __shared__ float hbuf[16][400];
    const int d = blockIdx.x;
    const float* xwd = xw + (long)d * 2048 * 1600;
    const float* wh  = whh + (long)d * 1600 * 400;
    const int lane = threadIdx.x & 31;
    const int wave = threadIdx.x >> 5;    // 0..24
    const int l16  = lane & 15;
    const int hl   = lane >> 4;
    const int col  = wave * 16 + l16;     // hidden unit column [0,400)

    for (int i = threadIdx.x; i < 16 * 400; i += blockDim.x)
        ((float*)hbuf)[i] = 0.f;
    v8f cstate = {0.f, 0.f, 0.f, 0.f, 0.f, 0.f, 0.f, 0.f};
    __syncthreads();

    // v2f views: (row*400 + k) is even for even k, rows are 8B aligned
    const v2f* hv   = (const v2f*)&hbuf[l16][0];
    const v2f* w0v  = (const v2f*)(wh + (long)(   0 + col) * 400 + 2 * hl);
    const v2f* w1v  = (const v2f*)(wh + (long)( 400 + col) * 400 + 2 * hl);
    const v2f* w2v  = (const v2f*)(wh + (long)( 800 + col) * 400 + 2 * hl);
    const v2f* w3v  = (const v2f*)(wh + (long)(1200 + col) * 400 + 2 * hl);

    for (int step = 0; step < 128; ++step) {
        const int t = d ? (127 - step) : step;
        v8f acc0 = {0.f,0.f,0.f,0.f,0.f,0.f,0.f,0.f};
        v8f acc1 = acc0, acc2 = acc0, acc3 = acc0;
        for (int k0 = 0; k0 < 400; k0 += 4) {
            const int kv = (k0 >> 1) + hl;     // hl folds the +2*hl float offset
            v2f a  = hv[kv];
            const int kw = k0 >> 1;            // w*v already offset by 2*hl
            v2f b0 = w0v[kw], b1 = w1v[kw], b2 = w2v[kw], b3 = w3v[kw];
            acc0 = __builtin_amdgcn_wmma_f32_16x16x4_f32(false, a, false, b0, (short)0, acc0, false, false);
            acc1 = __builtin_amdgcn_wmma_f32_16x16x4_f32(false, a, false, b1, (short)0, acc1, false, false);
            acc2 = __builtin_amdgcn_wmma_f32_16x16x4_f32(false, a, false, b2, (short)0, acc2, false, false);
            acc3 = __builtin_amdgcn_wmma_f32_16x16x4_f32(false, a, false, b3, (short)0, acc3, false, false);
        }
        __syncthreads();   // all waves done reading previous h
        for (int r = 0; r < 8; ++r) {
            const int m = r + 8 * hl;                       // batch row 0..15
            const long base = ((long)m * 128 + t) * 1600;
            const float gi = acc0[r] + xwd[base +    0 + col];
            const float gf = acc1[r] + xwd[base +  400 + col];
            const float gg = acc2[r] + xwd[base +  800 + col];
            const float go = acc3[r] + xwd[base + 1200 + col];
            const float c  = sigf(gf) * cstate[r] + sigf(gi) * tanhf(gg);
            const float h  = sigf(go) * tanhf(c);
            cstate[r] = c;
            hbuf[m][col] = h;
            out[((long)m * 128 + t) * 800 + d * 400 + col] = h;
        }
        __syncthreads();   // new h visible to all waves
    }
}

// ---------------------------------------------------------------------------
// x = concat(hf[:, :-1], hb[:, 1:])  ->  xcat[2032, 800]
// ---------------------------------------------------------------------------
__global__ void xcat_kernel(const float* __restrict__ h, float* __restrict__ xc)
{
    long i = (long)blockIdx.x * blockDim.x + threadIdx.x;
    if (i >= 2032L * 800) return;
    long r = i / 800, c = i % 800;
    long b = r / 127, t = r % 127;
    long src = (c < 400) ? ((b * 128 + t) * 800 + c)
                         : ((b * 128 + t + 1) * 800 + c);
    xc[i] = h[src];
}

__global__ void fill_col(float* __restrict__ p, int ld, int col)
{
    int r = blockIdx.x * blockDim.x + threadIdx.x;
    if (r < 2032) p[(long)r * ld + col] = 1.f;
}

// ---------------------------------------------------------------------------
// Inside-outside on a 127x127 chart; one block per batch element.
// Upper-triangular packing: 8001 entries per chart. score & working (alpha,
// then beta) charts in LDS (2 x 32 KB); alpha spilled to global for outside.
// Writes marg = mu * softmax(s_span) into d_out channels 0..1.
// ---------------------------------------------------------------------------
__device__ __forceinline__ int triidx(int i, int j) { return i * (253 - i) / 2 + (j - i - 1); }

__global__ void inside_outside(const float* __restrict__ sspan,
                               float* __restrict__ alphaG,
                               float* __restrict__ outp)
{
    constexpr int n = 127;
    constexpr int NT = 8001;
    __shared__ float scoreT[NT];
    __shared__ float wtri[NT];     // alpha during inside, beta during outside
    const int b = blockIdx.x;
    const int tid = threadIdx.x;
    const int bd = blockDim.x;
    const float* sp = sspan + (long)b * n * n * 2;
    float* aG = alphaG + (long)b * NT;

    // score = logsumexp over the 2 span channels
    for (int p = tid; p < n * n; p += bd) {
        int i = p / n, j = p % n;
        if (j > i) {
            float s0 = sp[p * 2], s1 = sp[p * 2 + 1];
            float m = fmaxf(s0, s1);
            scoreT[triidx(i, j)] = m + logf(expf(s0 - m) + expf(s1 - m));
        }
    }
    __syncthreads();

    // inside: alpha[i,j] = score[i,j] + LSE_k (alpha[i,k]+alpha[k,j])
    for (int w = 1; w < n; ++w) {
        for (int i = tid; i + w <= n - 1; i += bd) {
            const int j = i + w;
            float val = scoreT[triidx(i, j)];
            if (w > 1) {
                float mx = -3.402823e38f, s = 0.f;
                for (int k = i + 1; k < j; ++k) {
                    float t = wtri[triidx(i, k)] + wtri[triidx(k, j)];
                    if (t > mx) { s = s * expf(mx - t) + 1.f; mx = t; }
                    else        { s += expf(t - mx); }
                }
                val += mx + logf(s);
            }
            wtri[triidx(i, j)] = val;
        }
        __syncthreads();
    }

    // spill alpha, reuse wtri as beta
    for (int p = tid; p < NT; p += bd) aG[p] = wtri[p];
    __threadfence();
    __syncthreads();
    for (int p = tid; p < NT; p += bd) wtri[p] = 0.f;
    __syncthreads();
    if (tid == 0) wtri[triidx(0, n - 1)] = 1.f;
    __syncthreads();

    // outside (gather form): beta[i,j] = sum over parents, widths descending
    for (int w = n - 2; w >= 1; --w) {
        for (int i = tid; i + w <= n - 1; i += bd) {
            const int j = i + w;
            const float aij = aG[triidx(i, j)];
            float sum = 0.f;
            for (int j2 = j + 1; j2 <= n - 1; ++j2) {       // parent (i, j2), split at j
                const int pidx = triidx(i, j2);
                sum += wtri[pidx] * expf(aij + aG[triidx(j, j2)] + scoreT[pidx] - aG[pidx]);
            }
            for (int i2 = 0; i2 < i; ++i2) {                // parent (i2, j), split at i
                const int pidx = triidx(i2, j);
                sum += wtri[pidx] * expf(aG[triidx(i2, i)] + aij + scoreT[pidx] - aG[pidx]);
            }
            wtri[triidx(i, j)] = sum;
        }
        __syncthreads();
    }

    // marg[b,i,j,o] = mu[i,j] * softmax_o(s_span[b,i,j,:]) ; zero below diagonal
    for (int p = tid; p < n * n; p += bd) {
        int i = p / n, j = p % n;
        float* o = outp + ((long)b * n * n + p) * 130;
        if (j > i) {
            float mu = wtri[triidx(i, j)];
            float s0 = sp[p * 2], s1 = sp[p * 2 + 1];
            float m = fmaxf(s0, s1);
            float e0 = expf(s0 - m), e1 = expf(s1 - m);
            float den = e0 + e1;
            o[0] = mu * e0 / den;
            o[1] = mu * e1 / den;
        } else {
            o[0] = 0.f; o[1] = 0.f;
        }
    }
}

// ---------------------------------------------------------------------------
// Host side
// ---------------------------------------------------------------------------
static void launch_gemm(hipStream_t s, const float* A, const float* B, const float* bias,
                        float* C, const float* Wc,
                        int M, int N, int K, int lda, int ldb, int ldc, int stc,
                        long aSb, long aSo, long bSb, long bSo, long cSb, long cSo,
                        int biasSo, int flags, int oBase, int nb, int no)
{
    int tiles = ((M + 15) / 16) * ((N + 15) / 16);
    dim3 grid((tiles + 7) / 8, nb, no);
    gemm_wmma<<<grid, 256, 0, s>>>(A, B, bias, C, Wc, M, N, K, lda, ldb, ldc, stc,
                                   aSb, aSo, bSb, bSo, cSb, cSo, biasSo, flags, oBase);
}

extern "C" void kernel_launch(void* const* d_in, const int* in_sizes, int n_in,
                              void* d_out, int out_size, void* d_ws, size_t ws_size,
                              hipStream_t stream)
{
    const float* word_embed = (const float*)d_in[0];
    const float* w_ih_l0    = (const float*)d_in[1];
    const float* w_hh_l0    = (const float*)d_in[2];
    const float* b_l0       = (const float*)d_in[3];
    const float* w_ih       = (const float*)d_in[4];
    const float* w_hh       = (const float*)d_in[5];
    const float* b_lstm     = (const float*)d_in[6];
    const float* W_sl       = (const float*)d_in[7];
    const float* b_sl       = (const float*)d_in[8];
    const float* W_sr       = (const float*)d_in[9];
    const float* b_sr       = (const float*)d_in[10];
    const float* W_ll       = (const float*)d_in[11];
    const float* b_ll       = (const float*)d_in[12];
    const float* W_lr       = (const float*)d_in[13];
    const float* b_lr       = (const float*)d_in[14];
    const float* W_span     = (const float*)d_in[15];
    const float* W_label    = (const float*)d_in[16];
    const float* W_cluster  = (const float*)d_in[17];
    const int*   words      = (const int*)d_in[18];
    float* out = (float*)d_out;

    float* w = (float*)d_ws;
    float* emb    = w; w += 2048L * 300;
    float* bufA   = w; w += 2048L * 800;
    float* bufB   = w; w += 2048L * 800;
    float* xw     = w; w += 2L * 2048 * 1600;
    float* xcat   = w; w += 2032L * 800;
    float* sl1    = w; w += 2032L * 501;
    float* spr    = w; w += 2032L * 500;
    float* ll1    = w; w += 2032L * 101;
    float* lr1    = w; w += 2032L * 101;
    float* Tsp    = w; w += 2L * 2032 * 500;
    float* ssp    = w; w += 2032L * 127 * 2;
    float* alphaG = w; w += 16L * 8001;
    float* U      = w; w += 8L * 2032 * 101;

    // embedding
    embed_kernel<<<(2048 * 300 + 255) / 256, 256, 0, stream>>>(word_embed, words, emb);

    // layer 0: xW = emb @ wih^T + b  (per dir), then recurrence
    launch_gemm(stream, emb, w_ih_l0, b_l0, xw, W_cluster,
                2048, 1600, 300, 300, 300, 1600, 1,
                0, 0, 0, 1600L * 300, 0, 2048L * 1600,
                1600, FLAG_TRANSB | FLAG_BIAS, 0, 1, 2);
    lstm_scan<<<2, 800, 0, stream>>>(xw, w_hh_l0, bufA);

    // layer 1
    launch_gemm(stream, bufA, w_ih, b_lstm, xw, W_cluster,
                2048, 1600, 800, 800, 800, 1600, 1,
                0, 0, 0, 1600L * 800, 0, 2048L * 1600,
                1600, FLAG_TRANSB | FLAG_BIAS, 0, 1, 2);
    lstm_scan<<<2, 800, 0, stream>>>(xw, w_hh, bufB);

    // layer 2
    launch_gemm(stream, bufB, w_ih + 2L * 1600 * 800, b_lstm + 2L * 1600, xw, W_cluster,
                2048, 1600, 800, 800, 800, 1600, 1,
                0, 0, 0, 1600L * 800, 0, 2048L * 1600,
                1600, FLAG_TRANSB | FLAG_BIAS, 0, 1, 2);
    lstm_scan<<<2, 800, 0, stream>>>(xw, w_hh + 2L * 1600 * 400, bufA);

    // boundary features
    xcat_kernel<<<(int)((2032L * 800 + 255) / 256), 256, 0, stream>>>(bufA, xcat);

    // MLPs (leaky ReLU 0.1); sl1/ll1/lr1 get an appended ones column
    launch_gemm(stream, xcat, W_sl, b_sl, sl1, W_cluster, 2032, 500, 800, 800, 800, 501, 1,
                0, 0, 0, 0, 0, 0, 0, FLAG_TRANSB | FLAG_BIAS | FLAG_LEAKY, 0, 1, 1);
    launch_gemm(stream, xcat, W_sr, b_sr, spr, W_cluster, 2032, 500, 800, 800, 800, 500, 1,
                0, 0, 0, 0, 0, 0, 0, FLAG_TRANSB | FLAG_BIAS | FLAG_LEAKY, 0, 1, 1);
    launch_gemm(stream, xcat, W_ll, b_ll, ll1, W_cluster, 2032, 100, 800, 800, 800, 101, 1,
                0, 0, 0, 0, 0, 0, 0, FLAG_TRANSB | FLAG_BIAS | FLAG_LEAKY, 0, 1, 1);
    launch_gemm(stream, xcat, W_lr, b_lr, lr1, W_cluster, 2032, 100, 800, 800, 800, 101, 1,
                0, 0, 0, 0, 0, 0, 0, FLAG_TRANSB | FLAG_BIAS | FLAG_LEAKY, 0, 1, 1);
    fill_col<<<(2032 + 255) / 256, 256, 0, stream>>>(sl1, 501, 500);
    fill_col<<<(2032 + 255) / 256, 256, 0, stream>>>(ll1, 101, 100);
    fill_col<<<(2032 + 255) / 256, 256, 0, stream>>>(lr1, 101, 100);

    // s_span stage 1: T[o] = sl1 @ W_span[o]   (K=501, B not transposed)
    launch_gemm(stream, sl1, W_span, b_sl, Tsp, W_cluster, 2032, 500, 501, 501, 500, 500, 1,
                0, 0, 0, 501L * 500, 0, 2032L * 500, 0, 0, 0, 1, 2);
    // s_span stage 2: s_span[b,x,y,o] = T[o][b] @ spr[b]^T
    launch_gemm(stream, Tsp, spr, b_sl, ssp, W_cluster, 127, 127, 500, 500, 500, 127 * 2, 2,
                127L * 500, 2032L * 500, 127L * 500, 0, 127L * 127 * 2, 1,
                0, FLAG_TRANSB, 0, 16, 2);

    // inside-outside marginals -> out channels 0..1
    inside_outside<<<16, 128, 0, stream>>>(ssp, alphaG, out);

    // s_label in chunks of 8 labels; stage 2 adds marg @ W_cluster^T epilogue
    for (int c = 0; c < 16; ++c) {
        launch_gemm(stream, ll1, W_label + (long)c * 8 * 101 * 101, b_ll, U, W_cluster,
                    2032, 101, 101, 101, 101, 101, 1,
                    0, 0, 0, 101L * 101, 0, 2032L * 101, 0, 0, 0, 1, 8);
        launch_gemm(stream, U, lr1, b_ll, out + 2 + c * 8, W_cluster,
                    127, 127, 101, 101, 101, 127 * 130, 130,
                    127L * 101, 2032L * 101, 127L * 101, 0, 127L * 127 * 130, 1,
                    0, FLAG_TRANSB | FLAG_CLUSTER, c * 8, 16, 8);
    }
}